// TransConv_1254130451193
// MI455X (gfx1250) — compile-verified
//
#include <hip/hip_runtime.h>
#include <hip/hip_bf16.h>

typedef __attribute__((ext_vector_type(2))) float        v2f;
typedef __attribute__((ext_vector_type(8))) float        v8f;
typedef __attribute__((ext_vector_type(4))) unsigned int v4u;
typedef __attribute__((ext_vector_type(4))) int          v4i;
typedef __attribute__((ext_vector_type(8))) int          v8i;

#define KCURV   1.0f
#define EPSC    1e-6f
#define LN_EPS  1e-5f

__device__ __forceinline__ float lrelu(float x) { return x > 0.f ? x : 0.01f * x; }

// full wave32 sum-reduction; result broadcast to all lanes
__device__ __forceinline__ float wred(float v) {
    #pragma unroll
    for (int m = 16; m > 0; m >>= 1) v += __shfl_xor(v, m, 32);
    return v;
}

// 16-lane half-wave reduction (lanes 0-15 and 16-31 reduce independently)
__device__ __forceinline__ float hred(float v) {
    #pragma unroll
    for (int m = 8; m > 0; m >>= 1) v += __shfl_xor(v, m, 32);
    return v;
}

// ---------------------------------------------------------------------------
// TDM: DMA a 2D f32 tile [tileH x tileW] from a row-major tensor
// [rowsTensor x width] (row stride strideElems) into LDS at ldsOff.
// OOB rows/cols read as zero (descriptor-level bounds), so no lane guards are
// needed downstream. Tracked by TENSORcnt.
// D# layout per CDNA5 ISA ch.8: group0 {count, lds_addr, global_addr, type=2},
// group1 {data_size=4B, tensor_dim0/1, tile_dim0/1, tensor_dim0_stride}.
// ---------------------------------------------------------------------------
__device__ __forceinline__ void tdm_load_2d_f32(unsigned ldsOff, const void* gptr,
                                                unsigned width, unsigned rowsTensor,
                                                unsigned tileW, unsigned tileH,
                                                unsigned strideElems)
{
    unsigned long long ga = (unsigned long long)gptr;
    v4u g0;
    g0[0] = 1u;                                                  // count=1, user mode
    g0[1] = ldsOff;                                              // lds_addr
    g0[2] = (unsigned)ga;                                        // global_addr[31:0]
    g0[3] = (unsigned)((ga >> 32) & 0x01ffffffu) | (2u << 30);   // global_addr[56:32], type=2
    v8i g1;
    g1[0] = (int)(2u << 16);                                     // data_size = 4B
    g1[1] = (int)((width & 0xffffu) << 16);                      // tensor_dim0[15:0]
    g1[2] = (int)(((width >> 16) & 0xffffu) | ((rowsTensor & 0xffffu) << 16));
    g1[3] = (int)(((rowsTensor >> 16) & 0xffffu) | ((tileW & 0xffffu) << 16)); // tile_dim0
    g1[4] = (int)(tileH & 0xffffu);                              // tile_dim1; tile_dim2=0
    g1[5] = (int)strideElems;                                    // tensor_dim0_stride[31:0]
    g1[6] = 0;
    g1[7] = 0;
    v4i z4 = {0, 0, 0, 0};
#if __clang_major__ >= 23
    v8i z8 = {0, 0, 0, 0, 0, 0, 0, 0};
    __builtin_amdgcn_tensor_load_to_lds(g0, g1, z4, z4, z8, 0);
#else
    __builtin_amdgcn_tensor_load_to_lds(g0, g1, z4, z4, 0);
#endif
}

// ---------------------------------------------------------------------------
// Kernel 1: s_fc[N,32] = h[N,128] @ W_fc[1:,:] + (W_fc[0,:] + b_fc)
// One wave per 16-node tile. A-tile staged to LDS by the Tensor Data Mover;
// WMMA f32 16x16x4, 2 column tiles, 32 K-steps, A fragments from LDS.
// ---------------------------------------------------------------------------
__global__ void fc_gemm(const float* __restrict__ h, const float* __restrict__ Wfc,
                        const float* __restrict__ bfc, float* __restrict__ sfc, int N)
{
    __shared__ float At[8 * 16 * 128];     // 64 KB: one 16x128 tile per wave
    int lane = threadIdx.x & 31;
    int wave = threadIdx.x >> 5;
    int tile = blockIdx.x * 8 + wave;
    int node0 = tile * 16;
    if (node0 >= N) return;
    int hi = lane >> 4;
    int l15 = lane & 15;

    // async DMA this wave's A tile into its private LDS slice
    unsigned ldsOff = (unsigned)(size_t)&At[wave * 2048];
    tdm_load_2d_f32(ldsOff, h + (size_t)node0 * 128,
                    /*width*/128, /*rowsTensor*/(unsigned)(N - node0),
                    /*tileW*/128, /*tileH*/16, /*stride*/128);
    __builtin_amdgcn_s_wait_tensorcnt(0);

    const float* arow = &At[wave * 2048 + l15 * 128];
    v8f c0 = {}, c1 = {};
    for (int k0 = 0; k0 < 128; k0 += 4) {
        int kk = k0 + (hi ? 2 : 0);
        v2f a;
        a.x = arow[kk];
        a.y = arow[kk + 1];
        const float* wr = Wfc + (size_t)(1 + kk) * 32;   // skip ones-row 0
        v2f b0, b1;
        b0.x = wr[l15];        b0.y = wr[32 + l15];
        b1.x = wr[16 + l15];   b1.y = wr[48 + l15];
        c0 = __builtin_amdgcn_wmma_f32_16x16x4_f32(false, a, false, b0, (short)0, c0, false, false);
        c1 = __builtin_amdgcn_wmma_f32_16x16x4_f32(false, a, false, b1, (short)0, c1, false, false);
    }
    float bias0 = Wfc[l15] + bfc[l15];            // ones-column weight acts as bias
    float bias1 = Wfc[16 + l15] + bfc[16 + l15];
    #pragma unroll
    for (int r = 0; r < 8; ++r) {
        int node = node0 + r + (hi ? 8 : 0);
        if (node < N) {
            sfc[(size_t)node * 32 + l15]      = c0[r] + bias0;
            sfc[(size_t)node * 32 + 16 + l15] = c1[r] + bias1;
        }
    }
}

// ---------------------------------------------------------------------------
// Kernel 2: wave per node. residual = hyp_point(s_fc); node_pad = [t, lrelu(LN(s))] pad 36.
// ---------------------------------------------------------------------------
__global__ void node_pointwise(const float* __restrict__ sfc, const float* __restrict__ g,
                               const float* __restrict__ b, float* __restrict__ res,
                               float* __restrict__ pad, int N)
{
    int lane = threadIdx.x & 31;
    int n = blockIdx.x * (blockDim.x >> 5) + (threadIdx.x >> 5);
    if (n >= N) return;
    float s = sfc[(size_t)n * 32 + lane];
    float t = sqrtf(wred(s * s) + KCURV);
    res[(size_t)n * 33 + 1 + lane] = s;
    if (lane == 0) res[(size_t)n * 33] = t;
    float mean = wred(s) * (1.f / 32.f);
    float dv = s - mean;
    float var = wred(dv * dv) * (1.f / 32.f);
    float y = dv * rsqrtf(var + LN_EPS) * g[lane] + b[lane];
    float l = lrelu(y);
    float tn = sqrtf(wred(l * l) + KCURV);
    float* pr = pad + (size_t)n * 36;
    pr[1 + lane] = l;
    if (lane == 0) pr[0] = tn;
    if (lane < 3) pr[33 + lane] = 0.f;   // K padding for WMMA
}

// ---------------------------------------------------------------------------
// Kernel 3: qkv[N,384] = node_pad[N,36] @ Wcat[36,384] (+bias).
// Block = one 16-node tile staged once via TDM; 8 waves x 3 column tiles.
// ---------------------------------------------------------------------------
__global__ void qkv_gemm(const float* __restrict__ pad,
                         const float* __restrict__ Wq, const float* __restrict__ bq,
                         const float* __restrict__ Wk, const float* __restrict__ bk,
                         const float* __restrict__ Wv, const float* __restrict__ bv,
                         float* __restrict__ qkv, int N)
{
    __shared__ float Pt[16 * 36];          // 2.25 KB shared A tile
    int lane = threadIdx.x & 31;
    int wave = threadIdx.x >> 5;
    int node0 = blockIdx.x * 16;
    if (node0 >= N) return;                // uniform over block
    if (wave == 0) {
        unsigned ldsOff = (unsigned)(size_t)&Pt[0];
        tdm_load_2d_f32(ldsOff, pad + (size_t)node0 * 36,
                        /*width*/36, /*rowsTensor*/(unsigned)(N - node0),
                        /*tileW*/36, /*tileH*/16, /*stride*/36);
        __builtin_amdgcn_s_wait_tensorcnt(0);
    }
    __syncthreads();

    int hi = lane >> 4;
    int l15 = lane & 15;
    const float* prow = &Pt[l15 * 36];
    for (int i = 0; i < 3; ++i) {
        int ct  = wave * 3 + i;          // 0..23
        int col = ct * 16 + l15;         // 0..383
        int sel = col >> 7;              // 0:q 1:k 2:v
        int head = (col >> 5) & 3;
        int d = col & 31;
        const float* Wm = sel == 0 ? Wq : (sel == 1 ? Wk : Wv);
        const float* bm = sel == 0 ? bq : (sel == 1 ? bk : bv);
        const float* wh = Wm + (size_t)head * 33 * 32;
        v8f c = {};
        for (int k0 = 0; k0 < 36; k0 += 4) {
            int kk = k0 + (hi ? 2 : 0);
            v2f a;
            a.x = prow[kk];
            a.y = prow[kk + 1];
            v2f b;
            b.x = (kk < 33)     ? wh[(size_t)kk * 32 + d]       : 0.f;
            b.y = (kk + 1 < 33) ? wh[(size_t)(kk + 1) * 32 + d] : 0.f;
            c = __builtin_amdgcn_wmma_f32_16x16x4_f32(false, a, false, b, (short)0, c, false, false);
        }
        float bias = bm[head * 32 + d];
        #pragma unroll
        for (int r = 0; r < 8; ++r) {
            int node = node0 + r + (hi ? 8 : 0);
            if (node < N) qkv[(size_t)node * 384 + col] = c[r] + bias;
        }
    }
}

// ---------------------------------------------------------------------------
// Kernel 4: ktv[h][32][32] = sum_e phi_k^T phi_v ; ksum[h][32] = sum_e phi_k.
// GEMM with K = E: each wave owns one head, 4 edges per WMMA step,
// four 16x16 f32 accumulators in registers, one atomic flush at the end.
// ---------------------------------------------------------------------------
__global__ void ktv_kernel(const float* __restrict__ qkv, const int* __restrict__ src,
                           const float* __restrict__ nsc, float* __restrict__ ktv,
                           float* __restrict__ ksum, int E)
{
    int lane = threadIdx.x & 31;
    int gw = (blockIdx.x * blockDim.x + threadIdx.x) >> 5;
    int head = gw & 3;
    int w = gw >> 2;
    int nW = (gridDim.x * blockDim.x) >> 7;   // waves per head
    int l15 = lane & 15;
    bool hi = lane >= 16;
    float inv = 1.f / fabsf(nsc[0]);
    v8f acc[2][2] = {};
    float ks = 0.f;
    for (int c = w; c * 4 < E; c += nW) {
        int e0 = c * 4;
        float pk[4], pv[4];
        #pragma unroll
        for (int e = 0; e < 4; ++e) {
            float phk = 0.f, phv = 0.f;
            int ei = e0 + e;
            if (ei < E) {                           // uniform across wave
                int s = src[ei];
                const float* row = qkv + (size_t)s * 384 + head * 32 + lane;
                float kk = row[128];
                float vv = row[256];
                float xk = (lrelu(kk) + EPSC) * inv;
                float n2 = wred(xk * xk);
                float xp = xk * xk * xk;             // POWER_K = 3
                float np2 = wred(xp * xp);
                phk = sqrtf(n2) * rsqrtf(np2) * xp;  // fp()
                phv = (lrelu(vv) + EPSC) * inv;
            }
            pk[e] = phk; pv[e] = phv;
            ks += phk;
        }
        v2f aK[2], bV[2];
        #pragma unroll
        for (int half = 0; half < 2; ++half) {
            int sl = half * 16 + l15;
            float k0v = __shfl(pk[0], sl, 32), k1v = __shfl(pk[1], sl, 32);
            float k2v = __shfl(pk[2], sl, 32), k3v = __shfl(pk[3], sl, 32);
            aK[half].x = hi ? k2v : k0v;
            aK[half].y = hi ? k3v : k1v;
            float v0 = __shfl(pv[0], sl, 32), v1 = __shfl(pv[1], sl, 32);
            float v2 = __shfl(pv[2], sl, 32), v3 = __shfl(pv[3], sl, 32);
            bV[half].x = hi ? v2 : v0;
            bV[half].y = hi ? v3 : v1;
        }
        #pragma unroll
        for (int mh = 0; mh < 2; ++mh)
            #pragma unroll
            for (int dh = 0; dh < 2; ++dh)
                acc[mh][dh] = __builtin_amdgcn_wmma_f32_16x16x4_f32(
                    false, aK[mh], false, bV[dh], (short)0, acc[mh][dh], false, false);
    }
    float* kt = ktv + head * 1024;
    #pragma unroll
    for (int mh = 0; mh < 2; ++mh)
        #pragma unroll
        for (int dh = 0; dh < 2; ++dh)
            #pragma unroll
            for (int r = 0; r < 8; ++r) {
                int m = mh * 16 + r + (hi ? 8 : 0);
                int d = dh * 16 + l15;
                atomicAdd(&kt[m * 32 + d], acc[mh][dh][r]);
            }
    atomicAdd(&ksum[head * 32 + lane], ks);
}

// ---------------------------------------------------------------------------
// Kernel 5: wave per 16 edges. phi_q -> per-wave LDS tile; numerator is a WMMA
// GEMM against LDS-resident ktv (A=phi_q 16x32, 8 K-steps, 2 col tiles, 4
// heads); per-edge denominators divide the accumulator rows; edge_out
// projection [t,attn]@W_ho is a second WMMA GEMM (K padded 33->36);
// segment-sum atomics straight from the C-tile layout.
// ---------------------------------------------------------------------------
__global__ void edge_kernel(const float* __restrict__ qkv, const int* __restrict__ dst,
                            const float* __restrict__ ktv, const float* __restrict__ ksum,
                            const float* __restrict__ Who, const float* __restrict__ bho,
                            const float* __restrict__ nsc,
                            float* __restrict__ seg, float* __restrict__ cnt, int E)
{
    __shared__ float sKtv[4 * 32 * 32];
    __shared__ float sKs[4 * 32];
    __shared__ float sW[33 * 32];
    __shared__ float sB[32];
    __shared__ float sPhi[8][16 * 32];   // per-wave phi_q tile [edge][dim]
    __shared__ float sDen[8][16];        // per-wave denominators
    __shared__ float sEin[8][16 * 36];   // per-wave [t, attn, pad] tile
    for (int i = threadIdx.x; i < 4096; i += blockDim.x) sKtv[i] = ktv[i];
    for (int i = threadIdx.x; i < 128;  i += blockDim.x) sKs[i]  = ksum[i];
    for (int i = threadIdx.x; i < 1056; i += blockDim.x) sW[i]   = Who[i];
    if (threadIdx.x < 32) sB[threadIdx.x] = bho[threadIdx.x];
    __syncthreads();

    int lane = threadIdx.x & 31;
    int wave = threadIdx.x >> 5;
    int l15 = lane & 15;
    bool hi = lane >= 16;
    int e0 = (blockIdx.x * 8 + wave) * 16;
    if (e0 >= E) return;
    float inv = 1.f / fabsf(nsc[0]);
    float* phi = sPhi[wave];
    float* den = sDen[wave];
    float* ein = sEin[wave];

    v8f acc0 = {}, acc1 = {};            // attn accumulation [16 edges x 32 d]
    for (int h = 0; h < 4; ++h) {
        // phase 1: per-edge phi_q (lane = dim), denominators
        for (int e = 0; e < 16; ++e) {
            int ei = e0 + e;             // uniform per wave
            float p = 0.f;
            if (ei < E) {
                int n = dst[ei];
                float qv = qkv[(size_t)n * 384 + h * 32 + lane];
                float x = (lrelu(qv) + EPSC) * inv;
                float n2 = wred(x * x);
                float xp = x * x * x;
                float np2 = wred(xp * xp);
                p = sqrtf(n2) * rsqrtf(np2) * xp;
            }
            phi[e * 32 + lane] = p;
            float dn = wred(p * sKs[h * 32 + lane]) + EPSC;
            if (lane == 0) den[e] = dn;
        }
        asm volatile("" ::: "memory");   // order same-wave LDS write -> read

        // phase 2: numerator = phi @ ktv[h]  (WMMA, A from LDS, B from LDS)
        v8f c0 = {}, c1 = {};
        for (int k0 = 0; k0 < 32; k0 += 4) {
            int kk = k0 + (hi ? 2 : 0);
            v2f a;
            a.x = phi[l15 * 32 + kk];
            a.y = phi[l15 * 32 + kk + 1];
            const float* kt = sKtv + h * 1024 + kk * 32;
            v2f b0, b1;
            b0.x = kt[l15];        b0.y = kt[32 + l15];
            b1.x = kt[16 + l15];   b1.y = kt[48 + l15];
            c0 = __builtin_amdgcn_wmma_f32_16x16x4_f32(false, a, false, b0, (short)0, c0, false, false);
            c1 = __builtin_amdgcn_wmma_f32_16x16x4_f32(false, a, false, b1, (short)0, c1, false, false);
        }
        // divide each edge-row by its denominator and accumulate over heads
        #pragma unroll
        for (int r = 0; r < 8; ++r) {
            float dn = den[r + (hi ? 8 : 0)];
            acc0[r] += c0[r] / dn;
            acc1[r] += c1[r] / dn;
        }
    }

    // attn = acc/4; build ein = [t, attn, 0,0,0] per edge in LDS
    #pragma unroll
    for (int r = 0; r < 8; ++r) {
        int er = r + (hi ? 8 : 0);
        float a0 = acc0[r] * 0.25f;
        float a1 = acc1[r] * 0.25f;
        float s = hred(a0 * a0 + a1 * a1);     // per-half row norm
        float t = sqrtf(s + KCURV);
        ein[er * 36 + 1 + l15]  = a0;          // dims 0-15
        ein[er * 36 + 17 + l15] = a1;          // dims 16-31
        if (l15 == 0) {
            ein[er * 36]      = t;
            ein[er * 36 + 33] = 0.f;
            ein[er * 36 + 34] = 0.f;
            ein[er * 36 + 35] = 0.f;
        }
    }
    asm volatile("" ::: "memory");

    // edge_out = hyp_point(ein @ W_ho + b_ho) via WMMA (9 K-steps, K pad 36)
    v8f o0 = {}, o1 = {};
    for (int k0 = 0; k0 < 36; k0 += 4) {
        int kk = k0 + (hi ? 2 : 0);
        v2f a;
        a.x = ein[l15 * 36 + kk];
        a.y = ein[l15 * 36 + kk + 1];
        v2f b0, b1;
        b0.x = (kk < 33)     ? sW[kk * 32 + l15]            : 0.f;
        b0.y = (kk + 1 < 33) ? sW[(kk + 1) * 32 + l15]      : 0.f;
        b1.x = (kk < 33)     ? sW[kk * 32 + 16 + l15]       : 0.f;
        b1.y = (kk + 1 < 33) ? sW[(kk + 1) * 32 + 16 + l15] : 0.f;
        o0 = __builtin_amdgcn_wmma_f32_16x16x4_f32(false, a, false, b0, (short)0, o0, false, false);
        o1 = __builtin_amdgcn_wmma_f32_16x16x4_f32(false, a, false, b1, (short)0, o1, false, false);
    }
    float bv0 = sB[l15], bv1 = sB[16 + l15];
    #pragma unroll
    for (int r = 0; r < 8; ++r) {
        int er = r + (hi ? 8 : 0);
        int ei = e0 + er;
        float ev0 = o0[r] + bv0;
        float ev1 = o1[r] + bv1;
        float te = sqrtf(hred(ev0 * ev0 + ev1 * ev1) + KCURV);
        if (ei < E) {
            int n = dst[ei];
            atomicAdd(&seg[(size_t)n * 33 + 1 + l15],  ev0);
            atomicAdd(&seg[(size_t)n * 33 + 17 + l15], ev1);
            if (l15 == 0) {
                atomicAdd(&seg[(size_t)n * 33], te);
                atomicAdd(&cnt[n], 1.f);
            }
        }
    }
}

// ---------------------------------------------------------------------------
// Kernel 6: wave per node. segment mean, Lorentz midpoint with residual,
// LayerNorm, hyp_point -> out[N,33].
// ---------------------------------------------------------------------------
__global__ void final_kernel(const float* __restrict__ seg, const float* __restrict__ cnt,
                             const float* __restrict__ res, const float* __restrict__ g,
                             const float* __restrict__ b, float* __restrict__ out, int N)
{
    int lane = threadIdx.x & 31;
    int n = blockIdx.x * (blockDim.x >> 5) + (threadIdx.x >> 5);
    if (n >= N) return;
    float c = cnt[n];
    float rc = c > 0.f ? 1.f / c : 0.f;
    float aggs = seg[(size_t)n * 33 + 1 + lane] * rc;
    float aggt = seg[(size_t)n * 33] * rc;
    float aves = 0.5f * (aggs + res[(size_t)n * 33 + 1 + lane]);
    float avet = 0.5f * (aggt + res[(size_t)n * 33]);
    float inner = -avet * avet + wred(aves * aves);
    float dn = sqrtf(fmaxf(fabsf(inner), EPSC));
    float hs = aves / dn;                        // sqrt(K) = 1
    float mean = wred(hs) * (1.f / 32.f);
    float dv = hs - mean;
    float var = wred(dv * dv) * (1.f / 32.f);
    float y = dv * rsqrtf(var + LN_EPS) * g[lane] + b[lane];
    float to = sqrtf(wred(y * y) + KCURV);
    out[(size_t)n * 33 + 1 + lane] = y;
    if (lane == 0) out[(size_t)n * 33] = to;
}

// ---------------------------------------------------------------------------
extern "C" void kernel_launch(void* const* d_in, const int* in_sizes, int n_in,
                              void* d_out, int out_size, void* d_ws, size_t ws_size,
                              hipStream_t stream)
{
    const float* h    = (const float*)d_in[0];
    const int*   src  = (const int*)d_in[1];
    const int*   dst  = (const int*)d_in[2];
    const float* Wfc  = (const float*)d_in[3];
    const float* bfc  = (const float*)d_in[4];
    const float* ln1g = (const float*)d_in[5];
    const float* ln1b = (const float*)d_in[6];
    const float* Wq   = (const float*)d_in[7];
    const float* bq   = (const float*)d_in[8];
    const float* Wk   = (const float*)d_in[9];
    const float* bk   = (const float*)d_in[10];
    const float* Wv   = (const float*)d_in[11];
    const float* bv   = (const float*)d_in[12];
    const float* Who  = (const float*)d_in[13];
    const float* bho  = (const float*)d_in[14];
    const float* ln2g = (const float*)d_in[15];
    const float* ln2b = (const float*)d_in[16];
    const float* nsc  = (const float*)d_in[17];

    int N = in_sizes[0] / 128;
    int E = in_sizes[1];

    float* ws = (float*)d_ws;
    size_t off = 0;
    float* sfc  = ws + off; off += (size_t)N * 32;
    float* res  = ws + off; off += (size_t)N * 33;
    float* pad  = ws + off; off += (size_t)N * 36;
    float* qkv  = ws + off; off += (size_t)N * 384;
    float* ktv  = ws + off; off += 4096;
    float* ksum = ws + off; off += 128;
    float* seg  = ws + off; off += (size_t)N * 33;
    float* cnt  = ws + off; off += (size_t)N;

    // zero the accumulators (ktv..cnt are contiguous); capture-safe memset node
    hipMemsetAsync(ktv, 0, (4096 + 128 + (size_t)N * 33 + (size_t)N) * sizeof(float), stream);

    int tilesN = (N + 15) / 16;
    fc_gemm<<<(tilesN + 7) / 8, 256, 0, stream>>>(h, Wfc, bfc, sfc, N);
    node_pointwise<<<(N + 7) / 8, 256, 0, stream>>>(sfc, ln1g, ln1b, res, pad, N);
    qkv_gemm<<<tilesN, 256, 0, stream>>>(pad, Wq, bq, Wk, bk, Wv, bv, qkv, N);
    ktv_kernel<<<256, 256, 0, stream>>>(qkv, src, nsc, ktv, ksum, E);
    int edgeWaves = (E + 15) / 16;
    edge_kernel<<<(edgeWaves + 7) / 8, 256, 0, stream>>>(qkv, dst, ktv, ksum, Who, bho, nsc, seg, cnt, E);
    final_kernel<<<(N + 7) / 8, 256, 0, stream>>>(seg, cnt, res, ln2g, ln2b, (float*)d_out, N);
}